// ExtractSplitPosition_13005160972620
// MI455X (gfx1250) — compile-verified
//
#include <hip/hip_runtime.h>
#include <hip/hip_bf16.h>
#include <stdint.h>

// ---------------------------------------------------------------------------
// 1-D greedy NMS (ExtractSplitPosition) for MI455X / gfx1250.
//
// Roofline: ~50MB traffic => ~2.2us @ 23.3 TB/s. Compute is made comparable by
// compacting the ~20% valid candidates before the 50-step serial NMS loop.
// CDNA5 paths used: async global->LDS copy (ASYNCcnt) to stage deltas while
// VALU computes sigmoids; wave32 shuffle reductions (ds_bpermute).
// ---------------------------------------------------------------------------

#define FW        1024     // feat_width = img_width / 16
#define PER_LANE  32       // FW / wave32
#define RPB       2        // rows (waves) per block -> 48KB LDS
#define NOUT      50
#define STRIDE_F  16.0f
#define THRESH    0.7f

__global__ __launch_bounds__(32 * RPB) void nms1d_kernel(
    const float* __restrict__ logits,   // [B, FW]
    const float* __restrict__ delta,    // [B, FW, 2]
    const int*   __restrict__ imgw_p,   // scalar
    float* __restrict__ out,            // [B*50*2] positions ++ [B*50] scores
    int B)
{
    __shared__ float  s_delta[RPB][FW * 2];  // async-staged raw deltas (8KB/row)
    __shared__ float2 s_cs[RPB][FW];         // compacted (cent, work-score)
    __shared__ float2 s_pos[RPB][FW];        // compacted (p0, p1)

    const int lane = threadIdx.x & 31;
    const int wid  = threadIdx.x >> 5;
    const int row  = blockIdx.x * RPB + wid;
    if (row >= B) return;                    // wave-uniform

    const float* dlt = delta  + (size_t)row * (FW * 2);
    const float* lg  = logits + (size_t)row * FW;
    float*  sm_d = s_delta[wid];
    float2* cs   = s_cs[wid];
    float2* pos  = s_pos[wid];

    // ---- stage this row's deltas (8KB) into LDS with the async copy engine.
    // GVS mode: mem = SGPR64 base + VGPR32 offset. LDS dest offset = low 32
    // bits of the generic pointer (LDS aperture keeps offset in addr[31:0]).
    {
        const uint32_t lds0 = (uint32_t)(uintptr_t)(void*)sm_d;
        #pragma unroll
        for (int k = 0; k < 16; ++k) {
            uint32_t byteoff = (uint32_t)(lane * 16 + k * 512);
            uint32_t ldsaddr = lds0 + byteoff;
            asm volatile("global_load_async_to_lds_b128 %0, %1, %2"
                         :: "v"(ldsaddr), "v"(byteoff), "s"(dlt)
                         : "memory");
        }
    }

    // ---- sigmoid scores from global while the async copy is in flight.
    float sc[PER_LANE];
    #pragma unroll
    for (int i = 0; i < PER_LANE; ++i) {
        float x = lg[i * 32 + lane];
        sc[i] = 1.0f / (1.0f + __expf(-x));
    }

    const float wmax = (float)imgw_p[0] - 1.0f;

    asm volatile("s_wait_asynccnt 0" ::: "memory");

    // ---- cent/pos/valid, stats, and index-ordered stream compaction.
    float nval = 0.0f;
    float vmin =  __builtin_inff();
    float vmax = -__builtin_inff();
    uint32_t base = 0;                       // wave-uniform running slot base
    #pragma unroll
    for (int i = 0; i < PER_LANE; ++i) {
        const int e = i * 32 + lane;
        float d0 = sm_d[e * 2 + 0];
        float d1 = sm_d[e * 2 + 1];
        float c  = ((float)e + 0.5f) * STRIDE_F;
        float p0 = fminf(fmaxf(fmaf(d0, STRIDE_F, c), 0.0f), wmax);
        float p1 = fminf(fmaxf(fmaf(d1, STRIDE_F, c), 0.0f), wmax);
        float ce = 0.5f * (p0 + p1);
        bool valid = (sc[i] >= THRESH);
        uint32_t bal = (uint32_t)__ballot(valid);
        if (valid) {
            nval += 1.0f;
            vmin = fminf(vmin, ce);
            vmax = fmaxf(vmax, ce);
            uint32_t slot = base + (uint32_t)__popc(bal & ((1u << lane) - 1u));
            cs[slot]  = make_float2(ce, sc[i]);
            pos[slot] = make_float2(p0, p1);
        }
        base += (uint32_t)__popc(bal);
    }

    // ---- wave32 reductions for n, vmin, vmax.
    #pragma unroll
    for (int off = 16; off; off >>= 1) {
        nval += __shfl_xor(nval, off, 32);
        vmin  = fminf(vmin, __shfl_xor(vmin, off, 32));
        vmax  = fmaxf(vmax, __shfl_xor(vmax, off, 32));
    }
    const int   nv  = (int)base;                               // uniform
    const float dth = 0.55f * (vmax - vmin) / (nval - 1.0f);   // may be NaN/inf

    const size_t orow    = (size_t)row * NOUT;
    float* out_pos = out;                                      // [B,50,2]
    float* out_sc  = out + (size_t)B * NOUT * 2;               // [B,50,1]

    const float NEGINF = -__builtin_inff();
    float selc = __builtin_nanf("");   // iter 0: |c-NaN|<=dth is false -> no suppression

    // ---- greedy NMS over compacted candidates: fused suppress + argmax.
    int it = 0;
    for (; it < NOUT; ++it) {
        float bval = NEGINF, bcent = 0.0f;
        int   bslot = 0;
        for (int s = lane; s < nv; s += 32) {
            float2 cv = cs[s];
            float v = cv.y;
            if (__builtin_fabsf(cv.x - selc) <= dth) {   // suppress prev pick's window
                v = NEGINF;
                cs[s].y = NEGINF;
            }
            if (v > bval) { bval = v; bcent = cv.x; bslot = s; }
        }
        #pragma unroll
        for (int off = 16; off; off >>= 1) {
            float ov = __shfl_xor(bval,  off, 32);
            float oc = __shfl_xor(bcent, off, 32);
            int   oi = __shfl_xor(bslot, off, 32);
            if (ov > bval || (ov == bval && oi < bslot)) {
                bval = ov; bcent = oc; bslot = oi;
            }
        }
        if (!(bval > NEGINF)) break;     // uniform: no candidates left
        selc = bcent;
        float2 pp = pos[bslot];          // uniform broadcast LDS read
        if (lane == 0) {
            out_pos[(orow + it) * 2 + 0] = pp.x;
            out_pos[(orow + it) * 2 + 1] = pp.y;
            out_sc[orow + it]            = bval;
        }
    }
    // ---- zero padding for exhausted iterations (pad_to_fixed_size).
    for (; it < NOUT; ++it) {
        if (lane == 0) {
            out_pos[(orow + it) * 2 + 0] = 0.0f;
            out_pos[(orow + it) * 2 + 1] = 0.0f;
            out_sc[orow + it]            = 0.0f;
        }
    }
}

extern "C" void kernel_launch(void* const* d_in, const int* in_sizes, int n_in,
                              void* d_out, int out_size, void* d_ws, size_t ws_size,
                              hipStream_t stream) {
    const float* logits = (const float*)d_in[0];   // [B, 1024]
    const float* delta  = (const float*)d_in[1];   // [B, 1024, 2]
    const int*   imgw   = (const int*)d_in[2];     // scalar 16384
    float* out = (float*)d_out;

    const int B = in_sizes[0] / FW;                // 4096
    dim3 grid((B + RPB - 1) / RPB);
    dim3 block(32 * RPB);
    nms1d_kernel<<<grid, block, 0, stream>>>(logits, delta, imgw, out, B);
}